// GCN_38165079392788
// MI455X (gfx1250) — compile-verified
//
#include <hip/hip_runtime.h>

typedef float v2f __attribute__((ext_vector_type(2)));
typedef float v8f __attribute__((ext_vector_type(8)));

#define FEAT 128
#define NCLS 16

// ---------------------------------------------------------------------------
// Zero the accumulator + degree region of the workspace (poisoned by harness).
// ---------------------------------------------------------------------------
__global__ void __launch_bounds__(256) sage_zero_kernel(float* __restrict__ p, int n) {
    int i = blockIdx.x * 256 + threadIdx.x;
    if (i < n) p[i] = 0.0f;
}

// ---------------------------------------------------------------------------
// Fused dual GEMM via V_WMMA_F32_16X16X4_F32 (wave32, one 16x16 tile / wave):
//   hl = x @ W_l^T   (written to workspace; aggregated over edges afterwards)
//   hr = x @ W_r^T   (written directly into d_out; added at finalize)
// A (16x4 f32): VGPR0 = K0|K2, VGPR1 = K1|K3 across half-waves -> each lane
// loads a contiguous float2 from its row.  B (4x16) = W^T with the symmetric
// layout -> each lane loads a contiguous float2 from W[n][k].
// ---------------------------------------------------------------------------
__global__ void __launch_bounds__(256) sage_gemm_wmma(
    const float* __restrict__ x,
    const float* __restrict__ Wl,
    const float* __restrict__ Wr,
    float* __restrict__ hl,
    float* __restrict__ hr,
    int n_nodes, int n_tiles)
{
    const int lane = threadIdx.x & 31;
    const int wave = threadIdx.x >> 5;
    const int tile = blockIdx.x * 8 + wave;
    if (tile >= n_tiles) return;            // whole wave exits together

    const int row_base = tile * 16;
    const int half = lane >> 4;             // 0: lanes 0-15, 1: lanes 16-31
    const int lm   = lane & 15;

    int arow = row_base + lm;
    if (arow >= n_nodes) arow = n_nodes - 1;   // tail safety (N%16==0 -> no-op)

    const float* __restrict__ xrow  = x  + (size_t)arow * FEAT;
    const float* __restrict__ wlrow = Wl + (size_t)lm   * FEAT;
    const float* __restrict__ wrrow = Wr + (size_t)lm   * FEAT;

    v8f cl = {};
    v8f cr = {};

    for (int k0 = 0; k0 < FEAT; k0 += 4) {
        const int ks = k0 + half * 2;
        v2f a  = *(const v2f*)(xrow  + ks);
        v2f bl = *(const v2f*)(wlrow + ks);
        v2f br = *(const v2f*)(wrrow + ks);
        // 8 args: (neg_a, A, neg_b, B, c_mod, C, reuse_a, reuse_b)
        cl = __builtin_amdgcn_wmma_f32_16x16x4_f32(false, a, false, bl,
                                                   (short)0, cl, false, false);
        cr = __builtin_amdgcn_wmma_f32_16x16x4_f32(false, a, false, br,
                                                   (short)0, cr, false, false);
    }

    // C/D layout: VGPR r -> M = r (lanes 0-15) / M = 8+r (lanes 16-31), N = lm.
#pragma unroll
    for (int r = 0; r < 8; ++r) {
        int row = row_base + half * 8 + r;
        if (row < n_nodes) {
            hl[row * NCLS + lm] = cl[r];
            hr[row * NCLS + lm] = cr[r];
        }
    }
}

// ---------------------------------------------------------------------------
// Edge scatter in class space: 16 lanes per edge, one f32 atomic each.
// All traffic is L2-resident (h_l is 6.4 MB, accum 6.4 MB).
// ---------------------------------------------------------------------------
__global__ void __launch_bounds__(256) sage_scatter(
    const int* __restrict__ ei,       // [2, E] row-major
    const float* __restrict__ hl,     // [N, 16]
    float* __restrict__ accum,        // [N, 16]
    float* __restrict__ deg,          // [N]
    int n_edges)
{
    int t = blockIdx.x * 256 + threadIdx.x;
    int e = t >> 4;
    int c = t & 15;
    if (e >= n_edges) return;
    int src = ei[e];
    int dst = ei[n_edges + e];
    float v = hl[src * NCLS + c];
    unsafeAtomicAdd(&accum[dst * NCLS + c], v);
    if (c == 0) unsafeAtomicAdd(&deg[dst], 1.0f);
}

// ---------------------------------------------------------------------------
// out = relu(accum / max(deg,1) + b_l + hr)   (hr already resident in d_out)
// ---------------------------------------------------------------------------
__global__ void __launch_bounds__(256) sage_finalize(
    float* __restrict__ out,
    const float* __restrict__ accum,
    const float* __restrict__ deg,
    const float* __restrict__ bl,
    int n_nodes)
{
    int t = blockIdx.x * 256 + threadIdx.x;
    int i = t >> 4;
    int c = t & 15;
    if (i >= n_nodes) return;
    float d = fmaxf(deg[i], 1.0f);
    float v = accum[t] / d + bl[c] + out[t];
    out[t] = fmaxf(v, 0.0f);
}

extern "C" void kernel_launch(void* const* d_in, const int* in_sizes, int n_in,
                              void* d_out, int out_size, void* d_ws, size_t ws_size,
                              hipStream_t stream) {
    const float* x  = (const float*)d_in[0];   // [N, 128]
    const int*   ei = (const int*)d_in[1];     // [2, E]
    const float* Wl = (const float*)d_in[2];   // [16, 128]
    const float* bl = (const float*)d_in[3];   // [16]
    const float* Wr = (const float*)d_in[4];   // [16, 128]
    float* out = (float*)d_out;                // [N, 16]

    const int n_nodes = in_sizes[0] / FEAT;
    const int n_edges = in_sizes[1] / 2;

    // Workspace layout: hl [N*16] | accum [N*16] | deg [N]   (~13.2 MB)
    float* hl    = (float*)d_ws;
    float* accum = hl + (size_t)n_nodes * NCLS;
    float* deg   = accum + (size_t)n_nodes * NCLS;

    // 1) zero accum+deg (contiguous: N*17 floats)
    int nz = n_nodes * (NCLS + 1);
    sage_zero_kernel<<<(nz + 255) / 256, 256, 0, stream>>>(accum, nz);

    // 2) WMMA projection (hl -> ws, hr -> d_out)
    int n_tiles = (n_nodes + 15) / 16;
    sage_gemm_wmma<<<(n_tiles + 7) / 8, 256, 0, stream>>>(x, Wl, Wr, hl, out,
                                                          n_nodes, n_tiles);

    // 3) edge scatter (16 lanes per edge)
    long tot = (long)n_edges * NCLS;
    sage_scatter<<<(int)((tot + 255) / 256), 256, 0, stream>>>(ei, hl, accum, deg,
                                                               n_edges);

    // 4) finalize: mean, bias, self term, relu
    int tot2 = n_nodes * NCLS;
    sage_finalize<<<(tot2 + 255) / 256, 256, 0, stream>>>(out, accum, deg, bl,
                                                          n_nodes);
}